// Splatter_70300024701469
// MI455X (gfx1250) — compile-verified
//
#include <hip/hip_runtime.h>
#include <math.h>

#define NG   2048
#define IMW  128
#define IMH  128
#define GSTR 12   // packed floats per gaussian: u v A B | C opa r g | b 0 0 0

typedef __attribute__((ext_vector_type(2))) float v2f;
typedef __attribute__((ext_vector_type(8))) float v8f;

#if __has_builtin(__builtin_amdgcn_global_load_async_to_lds_b128) && \
    __has_builtin(__builtin_amdgcn_s_wait_asynccnt)
#define HAVE_ASYNC_LDS 1
// The builtin expects pointers to a 16-byte int vector (the b128 payload):
// param0 = global-AS source, param1 = LDS-AS destination.
typedef int v4i_b128 __attribute__((vector_size(4 * sizeof(int))));
typedef __attribute__((address_space(1))) v4i_b128* gas_b128_p;
typedef __attribute__((address_space(3))) v4i_b128* las_b128_p;
#endif

__device__ __forceinline__ void quat_to_rot(float w, float x, float y, float z, float* R) {
  float n = sqrtf(w*w + x*x + y*y + z*z) + 1e-12f;
  float inv = 1.0f / n;
  w *= inv; x *= inv; y *= inv; z *= inv;
  R[0] = 1.f - 2.f*(y*y + z*z); R[1] = 2.f*(x*y - w*z);       R[2] = 2.f*(x*z + w*y);
  R[3] = 2.f*(x*y + w*z);       R[4] = 1.f - 2.f*(x*x + z*z); R[5] = 2.f*(y*z - w*x);
  R[6] = 2.f*(x*z - w*y);       R[7] = 2.f*(y*z + w*x);       R[8] = 1.f - 2.f*(x*x + y*y);
}

// ---------------- Phase 1: per-gaussian projection + 2D covariance ----------------
__global__ void splat_preprocess(const float* __restrict__ pos,
                                 const float* __restrict__ opa,
                                 const float* __restrict__ quat,
                                 const float* __restrict__ scale,
                                 const float* __restrict__ w2c_quat,
                                 const float* __restrict__ w2c_tran,
                                 float* __restrict__ depth,
                                 float* __restrict__ par /* SoA: 6 x NG */) {
  int i = blockIdx.x * blockDim.x + threadIdx.x;
  if (i >= NG) return;

  float R[9];
  quat_to_rot(w2c_quat[0], w2c_quat[1], w2c_quat[2], w2c_quat[3], R);

  float p0 = pos[3*i], p1 = pos[3*i+1], p2 = pos[3*i+2];
  float x = R[0]*p0 + R[1]*p1 + R[2]*p2 + w2c_tran[0];
  float y = R[3]*p0 + R[4]*p1 + R[5]*p2 + w2c_tran[1];
  float z = R[6]*p0 + R[7]*p1 + R[8]*p2 + w2c_tran[2];

  float dnorm = sqrtf(x*x + y*y + z*z);
  bool mask = z > 1.1f;                     // NEAR
  float zs = mask ? z : 1.0f;
  float u = x / zs, v = y / zs;
  mask = mask && (fabsf(u) < 0.64f) && (fabsf(v) < 0.64f);   // W/(2FX), H/(2FY)
  float iz  = 1.0f / zs;
  float iz2 = iz * iz;

  // gaussian 3D covariance: cov3d = (Rg*S)(Rg*S)^T
  float G[9];
  quat_to_rot(quat[4*i], quat[4*i+1], quat[4*i+2], quat[4*i+3], G);
  float s0 = scale[3*i], s1 = scale[3*i+1], s2 = scale[3*i+2];
  float RS0=G[0]*s0, RS1=G[1]*s1, RS2=G[2]*s2;
  float RS3=G[3]*s0, RS4=G[4]*s1, RS5=G[5]*s2;
  float RS6=G[6]*s0, RS7=G[7]*s1, RS8=G[8]*s2;
  float c00 = RS0*RS0+RS1*RS1+RS2*RS2;
  float c01 = RS0*RS3+RS1*RS4+RS2*RS5;
  float c02 = RS0*RS6+RS1*RS7+RS2*RS8;
  float c11 = RS3*RS3+RS4*RS4+RS5*RS5;
  float c12 = RS3*RS6+RS4*RS7+RS5*RS8;
  float c22 = RS6*RS6+RS7*RS7+RS8*RS8;

  // M = rows 0,1 of (J @ rot); J row0=(iz,0,-x*iz2), row1=(0,iz,-y*iz2)
  float M00 = iz*R[0] - x*iz2*R[6];
  float M01 = iz*R[1] - x*iz2*R[7];
  float M02 = iz*R[2] - x*iz2*R[8];
  float M10 = iz*R[3] - y*iz2*R[6];
  float M11 = iz*R[4] - y*iz2*R[7];
  float M12 = iz*R[5] - y*iz2*R[8];

  // cov2d (2x2 symmetric) = M cov3d M^T
  float t0 = M00*c00 + M01*c01 + M02*c02;
  float t1 = M00*c01 + M01*c11 + M02*c12;
  float t2 = M00*c02 + M01*c12 + M02*c22;
  float a  = t0*M00 + t1*M01 + t2*M02 + 1e-6f;
  float b  = t0*M10 + t1*M11 + t2*M12;
  float q0 = M10*c00 + M11*c01 + M12*c02;
  float q1 = M10*c01 + M11*c11 + M12*c12;
  float q2 = M10*c02 + M11*c12 + M12*c22;
  float c  = q0*M10 + q1*M11 + q2*M12 + 1e-6f;

  float inv_det = 1.0f / fmaxf(a*c - b*b, 1e-12f);
  float An = c * inv_det;
  float Bn = b * inv_det;
  float Cn = a * inv_det;

  depth[i]        = mask ? dnorm : __builtin_inff();
  par[0*NG + i]   = u;
  par[1*NG + i]   = v;
  par[2*NG + i]   = An;
  par[3*NG + i]   = Bn;
  par[4*NG + i]   = Cn;
  par[5*NG + i]   = mask ? opa[i] : 0.0f;   // fold mask into opacity
}

// ---------------- Phase 2: bitonic sort by depth + gather packed AoS ----------------
__global__ void splat_sort_gather(const float* __restrict__ depth,
                                  const float* __restrict__ par,
                                  const float* __restrict__ rgb,
                                  float* __restrict__ packed) {
  __shared__ float skey[NG];
  __shared__ int   sidx[NG];
  for (int i = threadIdx.x; i < NG; i += blockDim.x) { skey[i] = depth[i]; sidx[i] = i; }
  __syncthreads();
  for (int k = 2; k <= NG; k <<= 1) {
    for (int j = k >> 1; j > 0; j >>= 1) {
      for (int i = threadIdx.x; i < NG; i += blockDim.x) {
        int ixj = i ^ j;
        if (ixj > i) {
          bool up = ((i & k) == 0);
          float ki = skey[i], kj = skey[ixj];
          bool sw = up ? (ki > kj) : (ki < kj);
          if (sw) {
            skey[i] = kj; skey[ixj] = ki;
            int t = sidx[i]; sidx[i] = sidx[ixj]; sidx[ixj] = t;
          }
        }
      }
      __syncthreads();
    }
  }
  for (int s = threadIdx.x; s < NG; s += blockDim.x) {
    int g = sidx[s];
    float* o = packed + s * GSTR;
    o[0]  = par[0*NG + g];  // u
    o[1]  = par[1*NG + g];  // v
    o[2]  = par[2*NG + g];  // A
    o[3]  = par[3*NG + g];  // B
    o[4]  = par[4*NG + g];  // C
    o[5]  = par[5*NG + g];  // effective opacity (mask folded)
    o[6]  = rgb[3*g+0];
    o[7]  = rgb[3*g+1];
    o[8]  = rgb[3*g+2];
    o[9]  = 0.0f; o[10] = 0.0f; o[11] = 0.0f;
  }
}

// ---------------- Phase 3: per-pixel compositing, RGB accumulate via WMMA ----------------
// One wave = 16 consecutive pixels (rows M of the 16x16 WMMA tile).
// Chunk of 4 gaussians per iteration = K dimension of V_WMMA_F32_16X16X4_F32.
__device__ __forceinline__ float gauss_alpha(float pu, float pv,
                                             float4 g0, float4 g1) {
  // g0 = (u, v, A, B), g1 = (C, opa, r, g)
  float dx = pu - g0.x;
  float dy = pv - g0.y;
  float power = -0.5f * (g0.z * dx * dx + g1.x * dy * dy) + g0.w * dx * dy;
  power = fminf(power, 0.0f);
  float al = g1.y * __expf(power);
  return fminf(al, 0.99f);
}

__launch_bounds__(256)
__global__ void splat_composite(const float* __restrict__ packed,
                                float* __restrict__ out) {
  __shared__ __align__(16) float lds[NG * GSTR];   // 96 KB of 320 KB WGP LDS

  // Stage the full sorted-gaussian table into LDS.
#if defined(HAVE_ASYNC_LDS)
  // CDNA5 async path: GLOBAL_LOAD_ASYNC_TO_LDS_B128, memory -> LDS direct,
  // tracked by ASYNCcnt (no VGPR round-trip).
  for (int i = threadIdx.x; i < NG * GSTR / 4; i += blockDim.x) {
    __builtin_amdgcn_global_load_async_to_lds_b128(
        (gas_b128_p)(packed + i * 4),
        (las_b128_p)(lds + i * 4),
        /*offset=*/0, /*cpol=*/0);
  }
  __builtin_amdgcn_s_wait_asynccnt(0);
  __syncthreads();
#else
  {
    const float4* src = (const float4*)packed;
    float4* dst = (float4*)lds;
    for (int i = threadIdx.x; i < NG * GSTR / 4; i += blockDim.x) dst[i] = src[i];
  }
  __syncthreads();
#endif

  const int lane = threadIdx.x & 31;
  const int col  = lane & 15;        // WMMA column / pixel-within-tile
  const int half = lane >> 4;        // half-wave: gaussians {0,1} vs {2,3} of chunk
  const int wid  = (blockIdx.x * blockDim.x + threadIdx.x) >> 5;
  const int pbase = wid << 4;        // 16 consecutive pixels per wave
  const int p  = pbase + col;
  const int px = p & (IMW - 1);
  const int py = p >> 7;
  const float pu = ((float)px + 0.5f - 64.0f) * 0.01f;   // /FX
  const float pv = ((float)py + 0.5f - 64.0f) * 0.01f;   // /FY

  float T = 1.0f;
  v8f acc = {};

  for (int g0 = 0; g0 < NG; g0 += 4) {
    const float* q = lds + (g0 + 2 * half) * GSTR;
    float4 qa0 = *(const float4*)(q);        // u v A B     (gaussian K = 2*half)
    float4 qa1 = *(const float4*)(q + 4);    // C opa r g
    float4 qa2 = *(const float4*)(q + 8);    // b - - -
    float4 qb0 = *(const float4*)(q + 12);   // gaussian K = 2*half + 1
    float4 qb1 = *(const float4*)(q + 16);
    float4 qb2 = *(const float4*)(q + 20);

    float a0 = gauss_alpha(pu, pv, qa0, qa1);
    float a1 = gauss_alpha(pu, pv, qb0, qb1);
    float om0 = 1.0f - a0;
    float prod = om0 * (1.0f - a1);                 // (1-a)(1-b) for this half's pair
    float other = __shfl_xor(prod, 16, 32);         // the other half's pair product
    float prefix = half ? T * other : T;            // front-to-back transmittance entry
    float w0 = prefix * a0;                         // weight for K = 2*half
    float w1 = prefix * om0 * a1;                   // weight for K = 2*half + 1
    T *= prod * other;                              // both halves agree

    // B operand: rows K=2*half (VGPR0), K=2*half+1 (VGPR1); cols 0..2 = rgb, rest 0
    float b0 = (col == 0) ? qa1.z : (col == 1) ? qa1.w : (col == 2) ? qa2.x : 0.0f;
    float b1 = (col == 0) ? qb1.z : (col == 1) ? qb1.w : (col == 2) ? qb2.x : 0.0f;

    v2f Aop = {w0, w1};
    v2f Bop = {b0, b1};
    acc = __builtin_amdgcn_wmma_f32_16x16x4_f32(
        /*neg_a=*/false, Aop, /*neg_b=*/false, Bop,
        /*c_mod=*/(short)0, acc, /*reuse_a=*/false, /*reuse_b=*/false);
  }

  // C/D layout: lane -> column N=col; VGPR j -> row M = j + 8*half
  if (col < 3) {
    #pragma unroll
    for (int j = 0; j < 8; ++j) {
      int pix = pbase + j + half * 8;
      out[pix * 3 + col] = acc[j];
    }
  }
}

extern "C" void kernel_launch(void* const* d_in, const int* in_sizes, int n_in,
                              void* d_out, int out_size, void* d_ws, size_t ws_size,
                              hipStream_t stream) {
  const float* pos      = (const float*)d_in[0];
  const float* rgb      = (const float*)d_in[1];
  const float* opa      = (const float*)d_in[2];
  const float* quat     = (const float*)d_in[3];
  const float* scale    = (const float*)d_in[4];
  const float* w2c_quat = (const float*)d_in[5];
  const float* w2c_tran = (const float*)d_in[6];
  float* out = (float*)d_out;

  float* ws     = (float*)d_ws;
  float* depth  = ws;                    // NG
  float* par    = ws + NG;               // 6*NG (SoA)
  float* packed = ws + NG + 6 * NG;      // NG*GSTR

  splat_preprocess<<<NG / 256, 256, 0, stream>>>(pos, opa, quat, scale,
                                                 w2c_quat, w2c_tran, depth, par);
  splat_sort_gather<<<1, 1024, 0, stream>>>(depth, par, rgb, packed);
  // 16384 pixels / 16 per wave = 1024 waves = 32768 threads
  splat_composite<<<(IMW * IMH / 16 * 32) / 256, 256, 0, stream>>>(packed, out);
}